// SbSLayer_78391743087295
// MI455X (gfx1250) — compile-verified
//
#include <hip/hip_runtime.h>
#include <stdint.h>

// Problem constants (match reference setup_inputs exactly)
#define KW    5
#define CIN   32
#define HH    48
#define WW_   48
#define BB    8
#define OXD   44          // H - K + 1
#define TT    256
#define NN    32          // output states == wave32 width
#define PP    800         // CIN*K*K
#define WAVES 8           // waves per block
#define NPOS  (BB*OXD*OXD) // 15488 positions, == 1936 blocks * 8 waves

// --- wave32 lane gather via LDS-crossbar (ds_bpermute), no LDS memory touched ---
__device__ __forceinline__ float lane_gather(float v, int srcLane) {
    int iv = __builtin_bit_cast(int, v);
    int r  = __builtin_amdgcn_ds_bpermute(srcLane << 2, iv); // bits[6:2] used -> wraps mod 32
    return __builtin_bit_cast(float, r);
}

__global__ __launch_bounds__(256)
void sbs_kernel(const float* __restrict__ x,      // (B,Cin,H,W)
                const float* __restrict__ u,      // (B,OX,OY,T)
                const float* __restrict__ w,      // (P,N) row-major
                const float* __restrict__ hinit,  // (N)
                float* __restrict__ out)          // (B,N,OX,OY)
{
    // CDNA5: 320KB LDS per WGP -> keep full weight table resident.
    __shared__ float lds_w[PP * NN];        // 102400 B
    __shared__ float lds_cdf[WAVES][PP];    //  25600 B
    __shared__ int   lds_spk[WAVES][TT];    //   8192 B
    __shared__ float lds_eps[TT];           //   1024 B
    __shared__ float lds_c1[TT];            //   1024 B

    const int tid  = threadIdx.x;
    const int wave = tid >> 5;
    const int lane = tid & 31;

    // ---------------------------------------------------------------
    // Stage weights (P*N*4 = 102400 B) global->LDS with the gfx1250
    // ASYNC DMA path (GLOBAL_LOAD_ASYNC_TO_LDS_B128, ASYNCcnt-tracked).
    // This overlaps with all CDF construction + binary searches below;
    // we only s_wait_asynccnt right before the H-dynamics need weights.
    // GVS addressing: mem = SADDR(64b) + VADDR(32b offset).
    // ---------------------------------------------------------------
    {
        const uint64_t gbase   = (uint64_t)(uintptr_t)w;
        const uint32_t ldsbase = (uint32_t)(uintptr_t)&lds_w[0]; // low 32 bits = LDS byte addr
#pragma unroll
        for (int k = 0; k < 25; ++k) {                 // 25 iters * 256 thr * 16B = 102400B
            uint32_t off  = (uint32_t)(k * 256 + tid) * 16u;  // contiguous, coalesced 512B/wave
            uint32_t ldst = ldsbase + off;
            asm volatile("global_load_async_to_lds_b128 %0, %1, %2"
                         :: "v"(ldst), "v"(off), "s"(gbase) : "memory");
        }
    }

    // Per-timestep constants: eps_t = (t+1)^-1/2, c1 = 1/(1+eps). One thread per t.
    {
        float tf  = (float)(tid + 1);
        float eps = 1.0f / sqrtf(tf);
        lds_eps[tid] = eps;
        lds_c1[tid]  = 1.0f / (1.0f + eps);
    }

    // Position handled by this wave (exact tiling: 1936 blocks * 8 waves = 15488)
    const int pos = blockIdx.x * WAVES + wave;
    const int b   = pos / (OXD * OXD);
    const int r   = pos - b * (OXD * OXD);
    const int ox  = r / OXD;
    const int oy  = r - ox * OXD;

    // ---------------------------------------------------------------
    // Patch gather + CDF. Lane c owns channel c's 25 patch elements:
    // feature index p = c*25 + ky*5 + kx  ->  x[b, c, ox+ky, oy+kx].
    // Lane-local inclusive prefix, then wave-level Hillis-Steele scan
    // of the 25-sums via ds_bpermute, then normalized store to LDS.
    // ---------------------------------------------------------------
    const float* xp = x + (((size_t)b * CIN + lane) * HH + ox) * WW_ + oy;
    float v[25];
#pragma unroll
    for (int ky = 0; ky < KW; ++ky)
#pragma unroll
        for (int kx = 0; kx < KW; ++kx)
            v[ky * KW + kx] = xp[ky * WW_ + kx];
#pragma unroll
    for (int j = 1; j < 25; ++j) v[j] += v[j - 1];

    const float mysum = v[24];
    float sc = mysum;
#pragma unroll
    for (int d = 1; d < 32; d <<= 1) {          // inclusive scan across lanes
        float o = lane_gather(sc, lane - d);    // lane<d wraps; masked by predicate
        if (lane >= d) sc += o;
    }
    const float total = lane_gather(sc, 31);
    const float excl  = sc - mysum;
    const float rtot  = 1.0f / total;

    float* cw = &lds_cdf[wave][0];
#pragma unroll
    for (int j = 0; j < 25; ++j)
        cw[lane * 25 + j] = (excl + v[j]) * rtot; // lane*25 strides are coprime w/ 64 banks
    // (same-wave DS ordering + compiler dscnt waits make cw safely readable below)

    // ---------------------------------------------------------------
    // Inverse-CDF sampling: 256 searchsorted(side='left') per position,
    // 8 per lane, binary search over the 800-entry LDS CDF.
    // ---------------------------------------------------------------
    const float* up = u + (size_t)pos * TT;     // u[b,ox,oy,:] — pos enumerates (b,ox,oy)
#pragma unroll
    for (int i = 0; i < 8; ++i) {
        const float uu = up[i * 32 + lane];     // coalesced 128B per wave
        int lo = 0, hi = PP;
        while (lo < hi) {                       // first idx with cdf[idx] >= u
            int  mid = (lo + hi) >> 1;
            bool lt  = cw[mid] < uu;
            lo = lt ? mid + 1 : lo;
            hi = lt ? hi      : mid;
        }
        lds_spk[wave][i * 32 + lane] = (lo < PP - 1) ? lo : (PP - 1);
    }

    // Weights must now be resident: drain ASYNCcnt, then block barrier.
    asm volatile("s_wait_asynccnt 0" ::: "memory");
    __syncthreads();

    // ---------------------------------------------------------------
    // Sequential SbS H-dynamics: lane n holds h[n] (N == wave32).
    //   ht = h * w[s];  ht /= sum(ht);  h = (h + eps*ht)/(1+eps)
    // Folded: h = h*c1 + ht * (eps*c1/sum),  c1 = 1/(1+eps).
    // Reduction = 5-step ds_bpermute butterfly.
    // ---------------------------------------------------------------
    float h = hinit[lane];
    const int* spw = &lds_spk[wave][0];
#pragma unroll 4
    for (int t = 0; t < TT; ++t) {
        const int   s  = spw[t];                    // uniform LDS broadcast load
        const float wv = lds_w[s * NN + lane];      // conflict-free: 32 consecutive dwords
        const float ht = h * wv;
        float sum = ht;
#pragma unroll
        for (int m = 16; m >= 1; m >>= 1)
            sum += lane_gather(sum, lane ^ m);
        const float coef = lds_eps[t] * __builtin_amdgcn_rcpf(sum) * lds_c1[t];
        h = h * lds_c1[t] + ht * coef;
    }

    // out[b, n=lane, ox, oy]
    out[(((size_t)b * NN + lane) * OXD + ox) * OXD + oy] = h;
}

extern "C" void kernel_launch(void* const* d_in, const int* in_sizes, int n_in,
                              void* d_out, int out_size, void* d_ws, size_t ws_size,
                              hipStream_t stream) {
    const float* x     = (const float*)d_in[0];
    const float* u     = (const float*)d_in[1];
    const float* w     = (const float*)d_in[2];
    const float* hinit = (const float*)d_in[3];
    float* out = (float*)d_out;
    (void)in_sizes; (void)n_in; (void)out_size; (void)d_ws; (void)ws_size;

    dim3 grid(NPOS / WAVES);   // 1936 blocks, 8 wave32's each -> 15488 positions exactly
    dim3 block(256);
    sbs_kernel<<<grid, block, 0, stream>>>(x, u, w, hinit, out);
}